// MultiInnerProductDecoder_14044543058209
// MI455X (gfx1250) — compile-verified
//
#include <hip/hip_runtime.h>
#include <hip/hip_bf16.h>
#include <stdint.h>

// DistMult edge scorer for MI455X (gfx1250, wave32).
// Memory-bound gather kernel: z (51MB) + weight (0.5MB) live in the 192MB L2;
// HBM traffic is ~107MB -> ~5us floor at 23.3 TB/s. WMMA would waste 15/16 of
// its MACs on off-diagonal terms here, so the CDNA5 feature used is the Tensor
// Data Mover: per-block edge-index slices are staged into LDS as 2D tiles.

#define IN_DIM          128
#define EDGES_PER_BLOCK 32
#define BLOCK_THREADS   256   // 8 wave32s
#define EDGES_PER_WAVE  4

typedef __attribute__((ext_vector_type(4))) unsigned int u32x4;
typedef __attribute__((ext_vector_type(8))) int          i32x8;
typedef __attribute__((ext_vector_type(4))) int          i32x4;

// Build a D# for an up-to-2D tile of 8-byte elements and issue TENSOR_LOAD_TO_LDS.
// tile:   tile0 x tile1 elements (tile1==0 -> 1D tile)
// tensor: dim0 x dim1 elements, row stride = stride0 (element units)
__device__ __forceinline__ void tdm_load_i64_tile(uint32_t lds_off, uint64_t gaddr,
                                                  uint32_t dim0, uint32_t dim1,
                                                  uint32_t tile0, uint32_t tile1,
                                                  uint64_t stride0) {
#if __has_builtin(__builtin_amdgcn_tensor_load_to_lds)
  u32x4 g0;
  g0.x = 1u;                                     // count=1, user mode, no gather
  g0.y = lds_off;                                // LDS byte address
  g0.z = (uint32_t)gaddr;                        // global_addr[31:0]
  g0.w = ((uint32_t)(gaddr >> 32) & 0x1FFFFFFu)  // global_addr[56:32]
       | (2u << 30);                             // type=2 ("image")
  i32x8 g1;
  g1[0] = (int)(3u << 16);                                   // data_size=3 (8B)
  g1[1] = (int)((dim0 & 0xFFFFu) << 16);                     // tensor_dim0[15:0]
  g1[2] = (int)((dim0 >> 16) | ((dim1 & 0xFFFFu) << 16));    // dim0[31:16], dim1[15:0]
  g1[3] = (int)(((dim1 >> 16) & 0xFFFFu) | (tile0 << 16));   // dim1[31:16], tile_dim0
  g1[4] = (int)(tile1 & 0xFFFFu);                            // tile_dim1 (tile_dim2=0)
  g1[5] = (int)(uint32_t)(stride0 & 0xFFFFFFFFu);            // tensor_dim0_stride[31:0]
  g1[6] = (int)((uint32_t)(stride0 >> 32) & 0xFFFFu);        // stride[47:32]
  g1[7] = 0;
  const i32x4 zero4 = {0, 0, 0, 0};                          // groups 2/3 unused (<=2D)
  const i32x8 zero8 = {0, 0, 0, 0, 0, 0, 0, 0};
  // 6-arg toolchain form: (g0, g1, g2, g3, extra, cpol)
  __builtin_amdgcn_tensor_load_to_lds(g0, g1, zero4, zero4, zero8, /*cpol=*/0);
#endif
}

__device__ __forceinline__ uint32_t lds_offset_of(const void* p) {
  // Generic pointers into LDS carry the byte offset in their low 32 bits.
  return (uint32_t)(uintptr_t)p;
}

__global__ __launch_bounds__(BLOCK_THREADS)
void distmult_score_kernel(const float* __restrict__ z,
                           const long long* __restrict__ edge_index,  // [2, E]
                           const long long* __restrict__ edge_type,   // [E]
                           const float* __restrict__ weight,          // [ET, 128]
                           float* __restrict__ out,                   // [E]
                           int n_edges) {
  __shared__ alignas(16) long long s_idx[2 * EDGES_PER_BLOCK]; // row0: src, row1: dst
  __shared__ alignas(16) long long s_typ[EDGES_PER_BLOCK];
  __shared__ float s_score[EDGES_PER_BLOCK];

  const int e0   = blockIdx.x * EDGES_PER_BLOCK;
  const int lane = threadIdx.x & 31;
  const int wave = threadIdx.x >> 5;

  // Wave 0 drives the Tensor Data Mover: stage this block's index slices into LDS.
  if (wave == 0) {
    // edge_index tile: 32 x 2 i64, rows are [src|dst], row stride = n_edges elems.
    tdm_load_i64_tile(lds_offset_of(s_idx),
                      (uint64_t)(uintptr_t)edge_index + (uint64_t)e0 * 8ull,
                      (uint32_t)n_edges, 2u, EDGES_PER_BLOCK, 2u,
                      (uint64_t)(uint32_t)n_edges);
    // edge_type tile: 32 x 1 i64 (1D).
    tdm_load_i64_tile(lds_offset_of(s_typ),
                      (uint64_t)(uintptr_t)edge_type + (uint64_t)e0 * 8ull,
                      (uint32_t)n_edges, 1u, EDGES_PER_BLOCK, 0u,
                      (uint64_t)(uint32_t)n_edges);
#if __has_builtin(__builtin_amdgcn_s_wait_tensorcnt)
    __builtin_amdgcn_s_wait_tensorcnt(0);
#endif
  }
  __syncthreads();

  // One wave per edge: lane l covers dims [4l, 4l+3]; row gathers are fully
  // coalesced 512B bursts (global_load_b128 per lane) that hit L2.
  const int col = lane * 4;
#pragma unroll
  for (int i = 0; i < EDGES_PER_WAVE; ++i) {
    const int le = wave * EDGES_PER_WAVE + i;
    const long long src = s_idx[le];                      // wave-uniform LDS broadcast
    const long long dst = s_idx[EDGES_PER_BLOCK + le];
    const long long typ = s_typ[le];

    const float4 a = *(const float4*)(z + (size_t)src * IN_DIM + col);
    const float4 b = *(const float4*)(z + (size_t)dst * IN_DIM + col);
    const float4 w = *(const float4*)(weight + (size_t)typ * IN_DIM + col);

    float partial = a.x * b.x * w.x;
    partial = fmaf(a.y * b.y, w.y, partial);
    partial = fmaf(a.z * b.z, w.z, partial);
    partial = fmaf(a.w * b.w, w.w, partial);

    // wave32 butterfly reduction
#pragma unroll
    for (int m = 16; m >= 1; m >>= 1)
      partial += __shfl_xor(partial, m, 32);

    if (lane == 0)
      s_score[le] = 1.0f / (1.0f + __expf(-partial));
  }
  __syncthreads();

  // One coalesced 128B store per block.
  if (threadIdx.x < EDGES_PER_BLOCK) {
    const int e = e0 + threadIdx.x;
    if (e < n_edges) out[e] = s_score[threadIdx.x];
  }
}

extern "C" void kernel_launch(void* const* d_in, const int* in_sizes, int n_in,
                              void* d_out, int out_size, void* d_ws, size_t ws_size,
                              hipStream_t stream) {
  const float*     z          = (const float*)d_in[0];      // [N_DRUGS, 128] f32
  const long long* edge_index = (const long long*)d_in[1];  // [2, E] i64
  const long long* edge_type  = (const long long*)d_in[2];  // [E] i64
  const float*     weight     = (const float*)d_in[3];      // [NUM_ET, 128] f32
  float*           out        = (float*)d_out;              // [E] f32

  const int n_edges = in_sizes[2];
  const int blocks  = (n_edges + EDGES_PER_BLOCK - 1) / EDGES_PER_BLOCK;

  distmult_score_kernel<<<blocks, BLOCK_THREADS, 0, stream>>>(
      z, edge_index, edge_type, weight, out, n_edges);
}